// EEGGCNet_71923522339509
// MI455X (gfx1250) — compile-verified
//
#include <hip/hip_runtime.h>
#include <hip/hip_bf16.h>

// ---------------------------------------------------------------------------
// EEGGCNet forward for MI455X (gfx1250, wave32).
// Dominant FLOPs (Chebyshev L-matmuls, separable 128x128 conv) run on
// V_WMMA_F32_16X16X4_F32. A-operands live in registers; B-operands are
// stored K-contiguous (column-major LDS) so each fragment is one b64 load.
// ---------------------------------------------------------------------------

typedef float v2f __attribute__((ext_vector_type(2)));
typedef float v8f __attribute__((ext_vector_type(8)));

#define B_   64
#define F1_  8
#define D_   16
#define K_   5
#define CH_  64
#define T_   640
#define KL_  80
#define NC_  4
#define EPS_ 1e-5f

// ---- workspace layout (in floats) -----------------------------------------
#define WS_STATS     0            // 4096 floats reserved
#define S_BN1SUM     0
#define S_BN1SQ      8
#define S_BN1SCALE   16
#define S_BN1SHIFT   24
#define S_BN2SUM     32
#define S_BN2SQ      160
#define S_BN2SCALE   288
#define S_BN2SHIFT   416
#define S_BN3SUM     544
#define S_BN3SQ      672
#define S_BN3SCALE   800
#define S_BN3SHIFT   928
#define WS_Y1        4096                              // B*F1*CH*T = 20,971,520
#define WS_G         (WS_Y1 + (size_t)B_*F1_*CH_*T_)   // B*128*T   =  5,242,880
#define WS_H         (WS_G  + (size_t)B_*128*T_)       // B*128*160 =  1,310,720
#define WS_Y2        (WS_H  + (size_t)B_*128*160)      // B*128*160 =  1,310,720
#define WS_WT        (WS_Y2 + (size_t)B_*128*160)      // 128*16*128 =   262,144

// ---------------------------------------------------------------------------
__global__ void init_stats_kernel(float* stats) {
  int i = blockIdx.x * 256 + threadIdx.x;
  if (i < 4096) stats[i] = 0.f;
}

// ---------------------------------------------------------------------------
// Repack sep_w (O,I,1,J) -> Wt (O,J,I) so the WMMA A fragment (two adjacent
// i values) is a single 64-bit global load.
// ---------------------------------------------------------------------------
__global__ void repack_w_kernel(const float* __restrict__ W, float* __restrict__ Wt) {
  int idx = blockIdx.x * 256 + threadIdx.x;
  if (idx >= 128 * 128 * 16) return;
  int j = idx & 15, i = (idx >> 4) & 127, o = idx >> 11;
  Wt[((size_t)o * 16 + j) * 128 + i] = W[idx];
}

// ---------------------------------------------------------------------------
// conv1 (1 -> F1 temporal conv, SAME pad 39/40) + bias, accumulate BN1 stats.
// grid (CH, B), 256 threads. VALU FMA (M=8 too small for WMMA tiles).
// ---------------------------------------------------------------------------
__global__ __launch_bounds__(256) void conv1_kernel(
    const float* __restrict__ X, const float* __restrict__ w1,
    const float* __restrict__ b1, float* __restrict__ y1,
    float* bn1_sum, float* bn1_sq) {
  __shared__ float xs[T_ + KL_ - 1];   // 719
  __shared__ float wsh[F1_ * KL_];
  __shared__ float s1[F1_], s2[F1_];
  const int c = blockIdx.x, b = blockIdx.y, tid = threadIdx.x;
  if (tid < F1_) { s1[tid] = 0.f; s2[tid] = 0.f; }
  for (int i = tid; i < F1_ * KL_; i += 256) wsh[i] = w1[i];
  const float* xr = X + ((size_t)b * CH_ + c) * T_;
  for (int i = tid; i < T_ + KL_ - 1; i += 256) {
    int g = i - 39;
    xs[i] = (g >= 0 && g < T_) ? xr[g] : 0.f;
  }
  __syncthreads();
  float ls1[F1_], ls2[F1_];
#pragma unroll
  for (int f = 0; f < F1_; ++f) { ls1[f] = 0.f; ls2[f] = 0.f; }
  for (int t = tid; t < T_; t += 256) {
    float acc[F1_];
#pragma unroll
    for (int f = 0; f < F1_; ++f) acc[f] = 0.f;
    for (int k = 0; k < KL_; ++k) {
      float xv = xs[t + k];
#pragma unroll
      for (int f = 0; f < F1_; ++f) acc[f] += wsh[f * KL_ + k] * xv;
    }
#pragma unroll
    for (int f = 0; f < F1_; ++f) {
      float v = acc[f] + b1[f];
      y1[((((size_t)b * F1_ + f) * CH_) + c) * T_ + t] = v;
      ls1[f] += v; ls2[f] += v * v;
    }
  }
#pragma unroll
  for (int f = 0; f < F1_; ++f) { atomicAdd(&s1[f], ls1[f]); atomicAdd(&s2[f], ls2[f]); }
  __syncthreads();
  if (tid < F1_) { atomicAdd(&bn1_sum[tid], s1[tid]); atomicAdd(&bn1_sq[tid], s2[tid]); }
}

// ---------------------------------------------------------------------------
__global__ void bn_finalize_kernel(const float* sum, const float* sq,
                                   const float* gamma, const float* beta,
                                   float* scale, float* shift, int n, float count) {
  int i = blockIdx.x * blockDim.x + threadIdx.x;
  if (i < n) {
    float m = sum[i] / count;
    float v = sq[i] / count - m * m;
    float sc = gamma[i] * rsqrtf(v + EPS_);
    scale[i] = sc;
    shift[i] = beta[i] - m * sc;
  }
}

// ---------------------------------------------------------------------------
// Chebyshev recursion via V_WMMA_F32_16X16X4_F32.
// One block = one (b, f, t-chunk of 64). Z buffers (COLUMN-major: [t][ch],
// stride 68 == 4 mod 64 -> conflict-free wave32 b64 loads) ping-pong in LDS.
// 8 waves x 2 N-tiles cover the 4x4 tile grid of L(64x64) @ Z(64x64).
// A fragments (L rows, fixed mt) preloaded once into 16 v2f registers and
// reused across both tiles and all 4 recursion steps: inner loop is
// 2 ds_load_b64 + 2 wmma per K-step.
// ---------------------------------------------------------------------------
#define TT   64
#define ZS2  68        // column stride of Z (chans per column + pad)
#define LST  65        // L row stride

__global__ __launch_bounds__(256) void cheb_wmma_kernel(
    const float* __restrict__ y1, const float* __restrict__ Lm,
    const float* __restrict__ cheb_w, const float* __restrict__ cheb_b,
    const float* __restrict__ bn1_scale, const float* __restrict__ bn1_shift,
    float* __restrict__ g_out, float* bn2_sum, float* bn2_sq) {
  __shared__ float Ls[CH_ * LST];
  __shared__ float Za[TT * ZS2];     // Za[t * ZS2 + ch]
  __shared__ float Zb[TT * ZS2];
  __shared__ float zbar[K_][TT];
  __shared__ float gs[D_], gq[D_];

  const int tc = blockIdx.x, f = blockIdx.y, b = blockIdx.z;
  const int tid = threadIdx.x;

  for (int i = tid; i < CH_ * CH_; i += 256)
    Ls[(i >> 6) * LST + (i & 63)] = Lm[i];
  const float sc = bn1_scale[f], sh = bn1_shift[f];
  const float* src = y1 + (((size_t)b * F1_ + f) * CH_) * T_ + tc * TT;
  for (int i = tid; i < CH_ * TT; i += 256) {
    int r = i / TT, cc = i % TT;          // r = channel, cc = time col
    Za[cc * ZS2 + r] = src[(size_t)r * T_ + cc] * sc + sh;
  }
  for (int i = tid; i < K_ * TT; i += 256) zbar[i / TT][i % TT] = 0.f;
  if (tid < D_) { gs[tid] = 0.f; gq[tid] = 0.f; }
  __syncthreads();

  // zbar[0] = column (channel) sums of Z0 -- contiguous within a column
  {
    int cc = tid & 63, rg = tid >> 6;     // 4 groups x 16 channels
    float s = 0.f;
    for (int r = rg * 16; r < rg * 16 + 16; ++r) s += Za[cc * ZS2 + r];
    atomicAdd(&zbar[0][cc], s);
  }

  const int w = tid >> 5, lane = tid & 31, lm = lane & 15, lh = lane >> 4;
  const int mt = w & 3, nt0 = w >> 2;     // tiles (mt,nt0) and (mt,nt0+2)

  // preload A fragments (invariant across steps and tiles): 16 x v2f
  v2f areg[16];
  {
    const int m = mt * 16 + lm;
#pragma unroll
    for (int kk = 0; kk < 16; ++kk) {
      const int k0 = kk * 4 + 2 * lh;
      areg[kk].x = Ls[m * LST + k0];
      areg[kk].y = Ls[m * LST + k0 + 1];
    }
  }
  __syncthreads();

  const int n0 = nt0 * 16 + lm;   // time column of tile 0
  const int n1 = n0 + 32;         // time column of tile 1 (nt0+2)

  float* cur = Za;   // Z_cur
  float* oth = Zb;   // Z_prev (recursion writes Z_next in place here)
  for (int s = 1; s < K_; ++s) {
    v8f acc0 = {}, acc1 = {};
#pragma unroll
    for (int kk = 0; kk < 16; ++kk) {
      const int k0 = kk * 4 + 2 * lh;     // even -> 8B aligned
      v2f b0 = *(const v2f*)&cur[n0 * ZS2 + k0];   // one ds_load_b64
      v2f b1 = *(const v2f*)&cur[n1 * ZS2 + k0];
      acc0 = __builtin_amdgcn_wmma_f32_16x16x4_f32(
          false, areg[kk], false, b0, (short)0, acc0, false, false);
      acc1 = __builtin_amdgcn_wmma_f32_16x16x4_f32(
          false, areg[kk], false, b1, (short)0, acc1, false, false);
    }
    // epilogue: Z_next = 2*Y - Z_prev (or Y at step 1), column sums for zbar
    float cs0 = 0.f, cs1 = 0.f;
#pragma unroll
    for (int v = 0; v < 8; ++v) {
      const int m = mt * 16 + v + 8 * lh;
      float v0 = (s == 1) ? acc0[v] : (2.f * acc0[v] - oth[n0 * ZS2 + m]);
      float v1 = (s == 1) ? acc1[v] : (2.f * acc1[v] - oth[n1 * ZS2 + m]);
      oth[n0 * ZS2 + m] = v0;   // owner-lane RMW, no cross-lane hazard
      oth[n1 * ZS2 + m] = v1;
      cs0 += v0; cs1 += v1;
    }
    atomicAdd(&zbar[s][n0], cs0);
    atomicAdd(&zbar[s][n1], cs1);
    __syncthreads();
    float* t2 = cur; cur = oth; oth = t2;
  }

  // g[b, f*16+d, t] = cheb_b[d] + sum_k mean_k[t] * cheb_w[k,d]; BN2 stats
  for (int p = tid; p < D_ * TT; p += 256) {
    const int d = p / TT, t = p % TT;
    float acc = cheb_b[d];
#pragma unroll
    for (int k = 0; k < K_; ++k)
      acc += zbar[k][t] * (1.f / 64.f) * cheb_w[k * D_ + d];
    g_out[(((size_t)b * 128) + f * D_ + d) * T_ + tc * TT + t] = acc;
    atomicAdd(&gs[d], acc);
    atomicAdd(&gq[d], acc * acc);
  }
  __syncthreads();
  if (tid < D_) {
    atomicAdd(&bn2_sum[f * D_ + tid], gs[tid]);
    atomicAdd(&bn2_sq[f * D_ + tid], gq[tid]);
  }
}

// ---------------------------------------------------------------------------
// BN2 + ELU + avgpool(4):  g(B,128,640) -> h(B,128,160)
// ---------------------------------------------------------------------------
__global__ void pool4_kernel(const float* __restrict__ g_in,
                             const float* __restrict__ sc,
                             const float* __restrict__ sh,
                             float* __restrict__ h) {
  int idx = blockIdx.x * 256 + threadIdx.x;
  if (idx >= B_ * 128 * 160) return;
  int t = idx % 160, c = (idx / 160) % 128, b = idx / (160 * 128);
  const float* src = g_in + ((size_t)b * 128 + c) * T_ + t * 4;
  float s = sc[c], s0 = sh[c], acc = 0.f;
#pragma unroll
  for (int r = 0; r < 4; ++r) {
    float v = src[r] * s + s0;
    acc += (v > 0.f) ? v : (__expf(v) - 1.f);
  }
  h[idx] = acc * 0.25f;
}

// ---------------------------------------------------------------------------
// Separable conv = 16 shifted 128x128 GEMMs, via WMMA f32 16x16x4.
// One block = one (b, t-chunk of 32). h slice (+halo 7/8) COLUMN-major in
// LDS (stride 132 == 4 mod 64 -> conflict-free b64 loads). A operand reads
// the repacked Wt (o,j,i) so each fragment is one global_load_b64; both
// N-tiles kept live so every A fragment feeds two wmma ops.
// ---------------------------------------------------------------------------
#define HS2 132        // column stride of hs (128 chans + pad)
__global__ __launch_bounds__(256) void sep_wmma_kernel(
    const float* __restrict__ h, const float* __restrict__ Wt,
    const float* __restrict__ sep_b, float* __restrict__ y2,
    float* bn3_sum, float* bn3_sq) {
  __shared__ float hs[48 * HS2];       // hs[t * HS2 + i]
  __shared__ float osum[128], osq[128];
  const int tc = blockIdx.x, b = blockIdx.y, tid = threadIdx.x;
  if (tid < 128) { osum[tid] = 0.f; osq[tid] = 0.f; }
  for (int i = tid; i < 128 * 48; i += 256) {
    int r = i / 48, x = i % 48;
    int gcol = tc * 32 - 7 + x;
    hs[x * HS2 + r] =
        (gcol >= 0 && gcol < 160) ? h[((size_t)b * 128 + r) * 160 + gcol] : 0.f;
  }
  __syncthreads();
  const int w = tid >> 5, lane = tid & 31, lm = lane & 15, lh = lane >> 4;
  const int mt = w;                 // o-tile
  const int o_a = mt * 16 + lm;     // A-operand row (fixed per lane)

  v8f acc0 = {}, acc1 = {};
  for (int j = 0; j < 16; ++j) {
    const float* Wrow = Wt + ((size_t)o_a * 16 + j) * 128;   // i contiguous
    __builtin_prefetch(Wrow + 128, 0, 0);                    // global_prefetch_b8
    const int c0 = lm + j;                                   // hpad col, nt=0
#pragma unroll 4
    for (int kk = 0; kk < 32; ++kk) {
      const int i0 = kk * 4 + 2 * lh;                        // even -> 8B aligned
      v2f a  = *(const v2f*)&Wrow[i0];                       // one b64 load
      v2f b0 = *(const v2f*)&hs[c0 * HS2 + i0];
      v2f b1 = *(const v2f*)&hs[(c0 + 16) * HS2 + i0];
      acc0 = __builtin_amdgcn_wmma_f32_16x16x4_f32(
          false, a, false, b0, (short)0, acc0, false, false);
      acc1 = __builtin_amdgcn_wmma_f32_16x16x4_f32(
          false, a, false, b1, (short)0, acc1, false, false);
    }
  }
#pragma unroll
  for (int v = 0; v < 8; ++v) {
    const int o = mt * 16 + v + 8 * lh;
    const int t0 = tc * 32 + lm;
    float v0 = acc0[v] + sep_b[o];
    float v1 = acc1[v] + sep_b[o];
    y2[((size_t)b * 128 + o) * 160 + t0]      = v0;
    y2[((size_t)b * 128 + o) * 160 + t0 + 16] = v1;
    atomicAdd(&osum[o], v0 + v1);
    atomicAdd(&osq[o], v0 * v0 + v1 * v1);
  }
  __syncthreads();
  if (tid < 128) {
    atomicAdd(&bn3_sum[tid], osum[tid]);
    atomicAdd(&bn3_sq[tid], osq[tid]);
  }
}

// ---------------------------------------------------------------------------
// BN3 + ELU + avgpool(8) + FC head. One block per batch element.
// ---------------------------------------------------------------------------
__global__ __launch_bounds__(256) void head_kernel(
    const float* __restrict__ y2, const float* __restrict__ sc3,
    const float* __restrict__ sh3, const float* __restrict__ fc_w,
    const float* __restrict__ fc_b, float* __restrict__ out) {
  __shared__ float p[128 * 20];
  __shared__ float os[NC_];
  const int b = blockIdx.x, tid = threadIdx.x;
  if (tid < NC_) os[tid] = 0.f;
  for (int i = tid; i < 128 * 20; i += 256) {
    int c = i / 20, tp = i % 20;
    const float* src = y2 + ((size_t)b * 128 + c) * 160 + tp * 8;
    float s = sc3[c], s0 = sh3[c], acc = 0.f;
#pragma unroll
    for (int r = 0; r < 8; ++r) {
      float v = src[r] * s + s0;
      acc += (v > 0.f) ? v : (__expf(v) - 1.f);
    }
    p[i] = acc * 0.125f;
  }
  __syncthreads();
  float a[NC_] = {0.f, 0.f, 0.f, 0.f};
  for (int i = tid; i < 128 * 20; i += 256) {
    float pv = p[i];
#pragma unroll
    for (int n = 0; n < NC_; ++n) a[n] += fc_w[n * 2560 + i] * pv;
  }
#pragma unroll
  for (int n = 0; n < NC_; ++n) atomicAdd(&os[n], a[n]);
  __syncthreads();
  if (tid < NC_) out[b * NC_ + tid] = os[tid] + fc_b[tid];
}

// ---------------------------------------------------------------------------
extern "C" void kernel_launch(void* const* d_in, const int* in_sizes, int n_in,
                              void* d_out, int out_size, void* d_ws, size_t ws_size,
                              hipStream_t stream) {
  const float* X       = (const float*)d_in[0];
  const float* L       = (const float*)d_in[1];
  const float* conv1_w = (const float*)d_in[2];
  const float* conv1_b = (const float*)d_in[3];
  const float* bn1_g   = (const float*)d_in[4];
  const float* bn1_b   = (const float*)d_in[5];
  const float* cheb_w  = (const float*)d_in[6];
  const float* cheb_b  = (const float*)d_in[7];
  const float* bn2_g   = (const float*)d_in[8];
  const float* bn2_b   = (const float*)d_in[9];
  const float* sep_w   = (const float*)d_in[10];
  const float* sep_b   = (const float*)d_in[11];
  const float* bn3_g   = (const float*)d_in[12];
  const float* bn3_b   = (const float*)d_in[13];
  const float* fc_w    = (const float*)d_in[14];
  const float* fc_b    = (const float*)d_in[15];
  float* out = (float*)d_out;

  float* ws  = (float*)d_ws;
  float* st  = ws + WS_STATS;
  float* y1  = ws + WS_Y1;
  float* g   = ws + WS_G;
  float* h   = ws + WS_H;
  float* y2  = ws + WS_Y2;
  float* wt  = ws + WS_WT;

  // 0) zero stat accumulators (ws is poisoned by the harness); repack sep_w
  init_stats_kernel<<<16, 256, 0, stream>>>(st);
  repack_w_kernel<<<(128 * 128 * 16 + 255) / 256, 256, 0, stream>>>(sep_w, wt);

  // 1) conv1 + BN1 stats
  conv1_kernel<<<dim3(CH_, B_), 256, 0, stream>>>(
      X, conv1_w, conv1_b, y1, st + S_BN1SUM, st + S_BN1SQ);

  // 2) BN1 finalize (count = B*CH*T per f-channel)
  bn_finalize_kernel<<<1, 32, 0, stream>>>(
      st + S_BN1SUM, st + S_BN1SQ, bn1_g, bn1_b,
      st + S_BN1SCALE, st + S_BN1SHIFT, F1_, (float)((size_t)B_ * CH_ * T_));

  // 3) Chebyshev recursion (WMMA) + cheb mix + BN2 stats
  cheb_wmma_kernel<<<dim3(T_ / TT, F1_, B_), 256, 0, stream>>>(
      y1, L, cheb_w, cheb_b, st + S_BN1SCALE, st + S_BN1SHIFT,
      g, st + S_BN2SUM, st + S_BN2SQ);

  // 4) BN2 finalize (count = B*T per channel)
  bn_finalize_kernel<<<1, 128, 0, stream>>>(
      st + S_BN2SUM, st + S_BN2SQ, bn2_g, bn2_b,
      st + S_BN2SCALE, st + S_BN2SHIFT, 128, (float)((size_t)B_ * T_));

  // 5) BN2 + ELU + pool4
  pool4_kernel<<<(B_ * 128 * 160 + 255) / 256, 256, 0, stream>>>(
      g, st + S_BN2SCALE, st + S_BN2SHIFT, h);

  // 6) separable conv (WMMA) + BN3 stats
  sep_wmma_kernel<<<dim3(160 / 32, B_), 256, 0, stream>>>(
      h, wt, sep_b, y2, st + S_BN3SUM, st + S_BN3SQ);

  // 7) BN3 finalize (count = B*160 per channel)
  bn_finalize_kernel<<<1, 128, 0, stream>>>(
      st + S_BN3SUM, st + S_BN3SQ, bn3_g, bn3_b,
      st + S_BN3SCALE, st + S_BN3SHIFT, 128, (float)((size_t)B_ * 160));

  // 8) BN3 + ELU + pool8 + FC
  head_kernel<<<B_, 256, 0, stream>>>(
      y2, st + S_BN3SCALE, st + S_BN3SHIFT, fc_w, fc_b, out);
}